// GaussianProcess_70257075028627
// MI455X (gfx1250) — compile-verified
//
#include <hip/hip_runtime.h>

// Additive RQ kernel: covar[i][j] = os * sum_d (1 + ((x[i][d]-x[j][d])/l[d])^2 / (2a))^(-a)
// mean = zeros(N).  N=2048, D=32, f32.
//
// Transcendental-bound (2 TRANS + ~5 VALU per (i,j,d) term). Strategy:
//  - symmetric: compute only upper-triangular 32x32 blocks, mirror-store (halves TRANS work)
//  - TDM tensor_load_to_lds stages x tiles into LDS (CDNA5 async-tensor path),
//    D# pad fields give LDS row stride 33 floats -> conflict-free ds_load_b32
//  - v_log_f32 / v_exp_f32 via builtins for the pow

#define DD   32   // feature dims
#define TILE 32   // block tile edge
#define LSTR 33   // LDS row stride in floats (32 data + 1 TDM pad dword)
#define NTHR 256

typedef __attribute__((ext_vector_type(4))) unsigned int tdm_u32x4;
typedef __attribute__((ext_vector_type(8))) int          tdm_i32x8;
typedef __attribute__((ext_vector_type(4))) int          tdm_i32x4;

// Issue one TDM load of a TILE x DD f32 tile (contiguous rows, row stride DD elems)
// from global memory into LDS at lds_off_bytes, padding 1 dword after every 32 dwords.
__device__ __forceinline__ void tdm_load_tile_32x32(unsigned lds_off_bytes,
                                                    const float* gsrc,
                                                    unsigned tensor_rows) {
  unsigned long long ga = (unsigned long long)gsrc;
  tdm_u32x4 g0;
  g0[0] = 1u;                                            // count=1 (valid user descriptor)
  g0[1] = lds_off_bytes;                                 // lds_addr [63:32]
  g0[2] = (unsigned)(ga & 0xFFFFFFFFull);                // global_addr lo [95:64]
  g0[3] = (unsigned)((ga >> 32) & 0x01FFFFFFull)         // global_addr hi [120:96]
        | (2u << 30);                                    // type=2 ("image") [127:126]
  tdm_i32x8 g1;
  g1[0] = (int)((2u << 16)                               // data_size = 4 bytes
              | (1u << 20)                               // pad_enable
              | (4u << 22));                             // pad_interval: 32 dwords
                                                         // pad_amount enc 0 = 1 dword
  g1[1] = (int)(((unsigned)DD) << 16);                   // tensor_dim0 lo16 @ [63:48]
  g1[2] = (int)((tensor_rows & 0xFFFFu) << 16);          // dim0 hi16=0 | tensor_dim1 lo16
  g1[3] = (int)(((tensor_rows >> 16) & 0xFFFFu)          // tensor_dim1 hi16
              | ((unsigned)TILE << 16));                 // tile_dim0 = 32 @ [127:112]
  g1[4] = (int)TILE;                                     // tile_dim1 = 32, tile_dim2 = 0
  g1[5] = (int)DD;                                       // tensor_dim0_stride lo32 = 32
  g1[6] = 0;                                             // stride hi / dim1_stride lo
  g1[7] = 0;
  tdm_i32x4 gz4 = {0, 0, 0, 0};                          // groups 2/3 unused (2D tensor)
  tdm_i32x8 gz8 = {0, 0, 0, 0, 0, 0, 0, 0};              // trailing group (clang-23 form)
  __builtin_amdgcn_tensor_load_to_lds(g0, g1, gz4, gz4, gz8, 0);
}

__global__ void __launch_bounds__(NTHR)
rq_additive_covar_kernel(const float* __restrict__ x,
                         const float* __restrict__ lengthscale,
                         const float* __restrict__ alpha_p,
                         const float* __restrict__ oscale_p,
                         float* __restrict__ covar, int n) {
  const int bi = blockIdx.y;   // row block
  const int bj = blockIdx.x;   // col block
  if (bi > bj) return;         // symmetric: upper-triangular blocks only

  __shared__ float xi_s[TILE * LSTR];
  __shared__ float xj_s[TILE * LSTR];

  const int t = (int)threadIdx.x;

  // Wave 0 issues the two TDM tile loads and waits on TENSORcnt.
  if ((t >> 5) == 0) {
    unsigned xi_off = (unsigned)(size_t)(__attribute__((address_space(3))) float*)xi_s;
    unsigned xj_off = (unsigned)(size_t)(__attribute__((address_space(3))) float*)xj_s;
    tdm_load_tile_32x32(xi_off, x + (size_t)bi * TILE * DD, (unsigned)n);
    tdm_load_tile_32x32(xj_off, x + (size_t)bj * TILE * DD, (unsigned)n);
    __builtin_amdgcn_s_wait_tensorcnt(0);
  }
  __syncthreads();

  // Normalize tiles in place: xs = x / lengthscale. 4 elems of each tile per thread.
  {
    const int e   = t * 4;            // 256 threads * 4 = 1024 = 32x32 elements
    const int row = e >> 5;
    const int col = e & 31;           // 4|32 so the 4 elems stay in one row
    const int idx = row * LSTR + col;
    const float l0 = lengthscale[col + 0];
    const float l1 = lengthscale[col + 1];
    const float l2 = lengthscale[col + 2];
    const float l3 = lengthscale[col + 3];
    xi_s[idx + 0] /= l0; xi_s[idx + 1] /= l1;
    xi_s[idx + 2] /= l2; xi_s[idx + 3] /= l3;
    xj_s[idx + 0] /= l0; xj_s[idx + 1] /= l1;
    xj_s[idx + 2] /= l2; xj_s[idx + 3] /= l3;
  }
  __syncthreads();

  const float alpha = alpha_p[0];
  const float os    = oscale_p[0];
  const float s     = 0.5f / alpha;   // 1/(2*alpha)
  const float na    = -alpha;

  // 2x2 register micro-tile per thread; 16x16 threads -> 32x32 outputs per block.
  const int tx = t & 15;
  const int ty = t >> 4;
  const int r0 = ty * 2, r1 = r0 + 1;   // rows within xi tile
  const int c0 = tx * 2, c1 = c0 + 1;   // rows within xj tile

  float acc00 = 0.f, acc01 = 0.f, acc10 = 0.f, acc11 = 0.f;

#pragma unroll
  for (int d = 0; d < DD; ++d) {
    const float a0 = xi_s[r0 * LSTR + d];
    const float a1 = xi_s[r1 * LSTR + d];
    const float b0 = xj_s[c0 * LSTR + d];
    const float b1 = xj_s[c1 * LSTR + d];

    const float d00 = a0 - b0, d01 = a0 - b1;
    const float d10 = a1 - b0, d11 = a1 - b1;

    const float u00 = __builtin_fmaf(d00 * d00, s, 1.0f);
    const float u01 = __builtin_fmaf(d01 * d01, s, 1.0f);
    const float u10 = __builtin_fmaf(d10 * d10, s, 1.0f);
    const float u11 = __builtin_fmaf(d11 * d11, s, 1.0f);

    // (u)^(-alpha) = exp2(-alpha * log2(u)); u >= 1 so log2 is safe.
    acc00 += __builtin_amdgcn_exp2f(na * __builtin_amdgcn_logf(u00));
    acc01 += __builtin_amdgcn_exp2f(na * __builtin_amdgcn_logf(u01));
    acc10 += __builtin_amdgcn_exp2f(na * __builtin_amdgcn_logf(u10));
    acc11 += __builtin_amdgcn_exp2f(na * __builtin_amdgcn_logf(u11));
  }

  const float v00 = acc00 * os, v01 = acc01 * os;
  const float v10 = acc10 * os, v11 = acc11 * os;

  const int gi0 = bi * TILE + r0, gi1 = gi0 + 1;
  const int gj0 = bj * TILE + c0, gj1 = gj0 + 1;

  // Direct tile (j contiguous): two b64 stores.
  *(float2*)&covar[(size_t)gi0 * n + gj0] = make_float2(v00, v01);
  *(float2*)&covar[(size_t)gi1 * n + gj0] = make_float2(v10, v11);
  // Mirrored tile (i contiguous): two b64 stores. For bi==bj these duplicate
  // bit-identical values ((a-b)^2 == (b-a)^2), which is benign.
  *(float2*)&covar[(size_t)gj0 * n + gi0] = make_float2(v00, v10);
  *(float2*)&covar[(size_t)gj1 * n + gi0] = make_float2(v01, v11);
}

__global__ void zero_mean_kernel(float* __restrict__ mean, int n) {
  const int i = blockIdx.x * blockDim.x + threadIdx.x;
  if (i < n) mean[i] = 0.0f;
}

extern "C" void kernel_launch(void* const* d_in, const int* in_sizes, int n_in,
                              void* d_out, int out_size, void* d_ws, size_t ws_size,
                              hipStream_t stream) {
  (void)n_in; (void)d_ws; (void)ws_size; (void)out_size;
  const float* x      = (const float*)d_in[0];
  const float* l      = (const float*)d_in[1];
  const float* alpha  = (const float*)d_in[2];
  const float* oscale = (const float*)d_in[3];

  const int n = in_sizes[0] / DD;     // 2048
  float* out   = (float*)d_out;
  float* mean  = out;                 // first n floats
  float* covar = out + n;             // then n*n floats

  zero_mean_kernel<<<(n + NTHR - 1) / NTHR, NTHR, 0, stream>>>(mean, n);

  dim3 grid(n / TILE, n / TILE);      // 64 x 64; lower-triangular blocks retire early
  rq_additive_covar_kernel<<<grid, NTHR, 0, stream>>>(x, l, alpha, oscale, covar, n);
}